// T3A_66726611911148
// MI455X (gfx1250) — compile-verified
//
#include <hip/hip_runtime.h>
#include <hip/hip_bf16.h>
#include <float.h>
#include <limits.h>

typedef __attribute__((ext_vector_type(2))) float v2f;
typedef __attribute__((ext_vector_type(8))) float v8f;

#define EPSN 1e-12f

// ---------------------------------------------------------------------------
// Wave/block reduction helpers (wave32)
// ---------------------------------------------------------------------------
__device__ __forceinline__ float wave_red_sum(float v) {
    for (int o = 16; o > 0; o >>= 1) v += __shfl_xor(v, o, 32);
    return v;
}

// 256-thread block sum. s8 must be shared float[8]. Returns the sum to all.
__device__ __forceinline__ float block_red_sum(float v, float* s8) {
    int lane = threadIdx.x & 31, w = threadIdx.x >> 5;
    v = wave_red_sum(v);
    if (lane == 0) s8[w] = v;
    __syncthreads();
    float r = 0.f;
    if (threadIdx.x < 32) {
        r = (threadIdx.x < 8) ? s8[threadIdx.x] : 0.f;
        for (int o = 4; o > 0; o >>= 1) r += __shfl_xor(r, o, 32);
        if (threadIdx.x == 0) s8[0] = r;
    }
    __syncthreads();
    r = s8[0];
    __syncthreads();   // protect s8 for a subsequent call
    return r;
}

// ---------------------------------------------------------------------------
// K1/K8: f32 WMMA GEMM  Out[M,N] = A[M,K] @ Bm[N,K]^T (+ bias[N])
// One wave per 16x16 output tile; V_WMMA_F32_16X16X4_F32, K stepped by 4.
// A-frag (16x4): lane<16 holds row=lane, K = k..k+1 ; lane>=16 K = k+2..k+3
// B-frag (4x16): lane<16 holds col=lane, same K split.
// D (16x16): vgpr v, lanes 0-15 -> M=v, lanes 16-31 -> M=v+8.
// ---------------------------------------------------------------------------
__global__ __launch_bounds__(32) void gemm_wmma_f32(
    const float* __restrict__ A, const float* __restrict__ Bm,
    const float* __restrict__ bias, float* __restrict__ Out,
    int M, int Nn, int Kd)
{
    const int lane = threadIdx.x;
    const int half = lane & 15;
    const int kOff = (lane >> 4) << 1;          // 0 or 2

    const int arow = blockIdx.y * 16 + half;
    const int bcol = blockIdx.x * 16 + half;
    const int arowc = arow < M  ? arow : M - 1; // clamp: keep EXEC full
    const int bcolc = bcol < Nn ? bcol : Nn - 1;

    const float* Ar = A  + (size_t)arowc * Kd + kOff;
    const float* Br = Bm + (size_t)bcolc * Kd + kOff;

    v8f acc = {};
    for (int k = 0; k < Kd; k += 4) {
        v2f a, b;
        a[0] = Ar[k]; a[1] = Ar[k + 1];
        b[0] = Br[k]; b[1] = Br[k + 1];
        acc = __builtin_amdgcn_wmma_f32_16x16x4_f32(
            false, a, false, b, (short)0, acc, false, false);
    }

    const int n     = blockIdx.x * 16 + half;
    const int mBase = blockIdx.y * 16 + ((lane >> 4) << 3);
    if (n < Nn) {
        const float bb = bias ? bias[n] : 0.f;
        for (int v = 0; v < 8; v++) {
            const int m = mBase + v;
            if (m < M) Out[(size_t)m * Nn + n] = acc[v] + bb;
        }
    }
}

// ---------------------------------------------------------------------------
// K2: per-batch-row softmax entropy + argmax from p[B,C]
// ---------------------------------------------------------------------------
__global__ __launch_bounds__(256) void ent_argmax_kernel(
    const float* __restrict__ p, float* __restrict__ E_all,
    int* __restrict__ y_all, int C, int N)
{
    __shared__ float s8[8];
    __shared__ float sv[8];
    __shared__ int   si[8];
    const int row = blockIdx.x;
    const float* pr = p + (size_t)row * C;
    const int tid = threadIdx.x, lane = tid & 31, w = tid >> 5;

    float bv = -FLT_MAX; int bi = INT_MAX;
    for (int c = tid; c < C; c += 256) {
        float v = pr[c];
        if (v > bv || (v == bv && c < bi)) { bv = v; bi = c; }
    }
    for (int o = 16; o > 0; o >>= 1) {
        float ov = __shfl_xor(bv, o, 32); int oi = __shfl_xor(bi, o, 32);
        if (ov > bv || (ov == bv && oi < bi)) { bv = ov; bi = oi; }
    }
    if (lane == 0) { sv[w] = bv; si[w] = bi; }
    __syncthreads();
    if (tid < 32) {
        float v2 = (tid < 8) ? sv[tid] : -FLT_MAX;
        int   i2 = (tid < 8) ? si[tid] : INT_MAX;
        for (int o = 4; o > 0; o >>= 1) {
            float ov = __shfl_xor(v2, o, 32); int oi = __shfl_xor(i2, o, 32);
            if (ov > v2 || (ov == v2 && oi < i2)) { v2 = ov; i2 = oi; }
        }
        if (tid == 0) { sv[0] = v2; si[0] = i2; }
    }
    __syncthreads();
    const float maxv = sv[0];
    const int amax = si[0];
    __syncthreads();

    float zz = 0.f, s1 = 0.f;
    for (int c = tid; c < C; c += 256) {
        float t = pr[c] - maxv;
        float e = expf(t);
        zz += e; s1 += t * e;
    }
    zz = block_red_sum(zz, s8);
    s1 = block_red_sum(s1, s8);
    if (tid == 0) {
        E_all[N + row] = logf(zz) - s1 / zz;   // softmax entropy
        y_all[N + row] = amax;
    }
}

// ---------------------------------------------------------------------------
// K3: support rows: class = argmax(one-hot labels row); copy ent -> E_all
// One wave per row, lanes stride over C.
// ---------------------------------------------------------------------------
__global__ __launch_bounds__(256) void support_argmax_kernel(
    const float* __restrict__ labels, const float* __restrict__ ent,
    int* __restrict__ y_all, float* __restrict__ E_all, int N, int C)
{
    const int row = blockIdx.x * 8 + (threadIdx.x >> 5);
    if (row >= N) return;
    const int lane = threadIdx.x & 31;
    const float* lr = labels + (size_t)row * C;
    float bv = -FLT_MAX; int bi = INT_MAX;
    for (int c = lane; c < C; c += 32) {
        float v = lr[c];
        if (v > bv || (v == bv && c < bi)) { bv = v; bi = c; }
    }
    for (int o = 16; o > 0; o >>= 1) {
        float ov = __shfl_xor(bv, o, 32); int oi = __shfl_xor(bi, o, 32);
        if (ov > bv || (ov == bv && oi < bi)) { bv = ov; bi = oi; }
    }
    if (lane == 0) { y_all[row] = bi; E_all[row] = ent[row]; }
}

// ---------------------------------------------------------------------------
// K4: class histogram (integer atomics -> deterministic)
// ---------------------------------------------------------------------------
__global__ void count_kernel(const int* __restrict__ y_all,
                             int* __restrict__ counts, int Ntot)
{
    int r = blockIdx.x * blockDim.x + threadIdx.x;
    if (r < Ntot) atomicAdd(&counts[y_all[r]], 1);
}

// K5: serial exclusive scan over C classes (C=1000, trivial)
__global__ void scan_kernel(const int* __restrict__ counts,
                            int* __restrict__ offsets, int C)
{
    if (blockIdx.x == 0 && threadIdx.x == 0) {
        int acc = 0;
        for (int c = 0; c < C; c++) { offsets[c] = acc; acc += counts[c]; }
        offsets[C] = acc;
    }
}

// ---------------------------------------------------------------------------
// K6: deterministic compaction: one wave per class; ballot/popcount keeps
// members in ascending original-row order (stable, no atomics).
// ---------------------------------------------------------------------------
__global__ __launch_bounds__(32) void compact_kernel(
    const int* __restrict__ y_all, const int* __restrict__ offsets,
    int* __restrict__ idxArr, int Ntot)
{
    const int c = blockIdx.x;
    const int lane = threadIdx.x;
    int base = offsets[c];
    for (int r0 = 0; r0 < Ntot; r0 += 32) {
        const int r = r0 + lane;
        const bool match = (r < Ntot) && (y_all[r] == c);
        unsigned long long m = __ballot(match);
        if (match) {
            int pos = base + __popcll(m & ((1ull << lane) - 1ull));
            idxArr[pos] = r;
        }
        base += __popcll(m);
    }
}

// ---------------------------------------------------------------------------
// K7: per-class top-K selection by lowest entropy, stable-sort semantics:
// keep i iff #{j: E_j < E_i  ||  (E_j == E_i && orig_j < orig_i)} < K
// ---------------------------------------------------------------------------
__global__ __launch_bounds__(256) void select_kernel(
    const int* __restrict__ offsets, const int* __restrict__ counts,
    const int* __restrict__ idxArr, const float* __restrict__ E_all,
    const int* __restrict__ fkp, int* __restrict__ keepM)
{
    __shared__ float sE[4096];
    __shared__ int   sO[4096];
    const int c = blockIdx.x;
    const int start = offsets[c];
    const int cnt = counts[c];
    const int K = *fkp;
    const int tid = threadIdx.x;

    if (cnt <= K) {
        for (int i = tid; i < cnt; i += 256) keepM[start + i] = 1;
        return;
    }
    const bool useLds = (cnt <= 4096);
    if (useLds) {
        for (int i = tid; i < cnt; i += 256) {
            int r = idxArr[start + i];
            sE[i] = E_all[r]; sO[i] = r;
        }
        __syncthreads();
    }
    for (int i = tid; i < cnt; i += 256) {
        float Ei; int Oi;
        if (useLds) { Ei = sE[i]; Oi = sO[i]; }
        else        { Oi = idxArr[start + i]; Ei = E_all[Oi]; }
        int rank = 0;
        for (int j = 0; j < cnt; j++) {
            float Ej; int Oj;
            if (useLds) { Ej = sE[j]; Oj = sO[j]; }
            else        { Oj = idxArr[start + j]; Ej = E_all[Oj]; }
            rank += (Ej < Ei) || (Ej == Ei && Oj < Oi);
        }
        keepM[start + i] = (rank < K) ? 1 : 0;
    }
}

// ---------------------------------------------------------------------------
// K8: per-class prototype: sum of L2-normalized kept rows, then
// column-normalize -> Wn[c, 0..D-1]. One 256-thread block per class,
// thread t owns d = t, t+256, t+512 (D <= 768).
// ---------------------------------------------------------------------------
__global__ __launch_bounds__(256) void proto_kernel(
    const int* __restrict__ offsets, const int* __restrict__ counts,
    const int* __restrict__ idxArr, const int* __restrict__ keepM,
    const float* __restrict__ supports, const float* __restrict__ z,
    float* __restrict__ Wn, int N, int D)
{
    __shared__ float s8[8];
    const int c = blockIdx.x;
    const int t = threadIdx.x;
    const int start = offsets[c], cnt = counts[c];
    const int d0 = t, d1 = t + 256, d2 = t + 512;

    float a0 = 0.f, a1 = 0.f, a2 = 0.f;
    for (int m = 0; m < cnt; m++) {
        if (!keepM[start + m]) continue;           // uniform across block
        const int r = idxArr[start + m];
        const float* row = (r < N) ? (supports + (size_t)r * D)
                                   : (z + (size_t)(r - N) * D);
        float v0 = (d0 < D) ? row[d0] : 0.f;
        float v1 = (d1 < D) ? row[d1] : 0.f;
        float v2 = (d2 < D) ? row[d2] : 0.f;
        float ss = block_red_sum(v0 * v0 + v1 * v1 + v2 * v2, s8);
        float sc = 1.0f / fmaxf(sqrtf(ss), EPSN);
        a0 += v0 * sc; a1 += v1 * sc; a2 += v2 * sc;
    }
    float css = block_red_sum(a0 * a0 + a1 * a1 + a2 * a2, s8);
    float sc = 1.0f / fmaxf(sqrtf(css), EPSN);
    if (d0 < D) Wn[(size_t)c * D + d0] = a0 * sc;
    if (d1 < D) Wn[(size_t)c * D + d1] = a1 * sc;
    if (d2 < D) Wn[(size_t)c * D + d2] = a2 * sc;
}

// ---------------------------------------------------------------------------
// Host orchestration
// ---------------------------------------------------------------------------
extern "C" void kernel_launch(void* const* d_in, const int* in_sizes, int n_in,
                              void* d_out, int out_size, void* d_ws, size_t ws_size,
                              hipStream_t stream) {
    const float* z        = (const float*)d_in[0];
    const float* W        = (const float*)d_in[1];
    const float* b        = (const float*)d_in[2];
    const float* supports = (const float*)d_in[3];
    const float* labels   = (const float*)d_in[4];
    const float* ent      = (const float*)d_in[5];
    const int*   fk       = (const int*)d_in[6];

    const int C    = in_sizes[2];
    const int D    = in_sizes[1] / C;
    const int B    = in_sizes[0] / D;
    const int N    = in_sizes[5];
    const int Ntot = N + B;

    // workspace carve-up
    char* w = (char*)d_ws;
    float* p      = (float*)w; w += sizeof(float) * (size_t)B * C;
    float* E_all  = (float*)w; w += sizeof(float) * (size_t)Ntot;
    int*   y_all  = (int*)w;   w += sizeof(int)   * (size_t)Ntot;
    int*   counts = (int*)w;   w += sizeof(int)   * (size_t)C;
    int*   offs   = (int*)w;   w += sizeof(int)   * (size_t)(C + 1);
    int*   idxArr = (int*)w;   w += sizeof(int)   * (size_t)Ntot;
    int*   keepM  = (int*)w;   w += sizeof(int)   * (size_t)Ntot;
    float* Wn     = (float*)w; w += sizeof(float) * (size_t)C * D;

    const dim3 gGemm((C + 15) / 16, (B + 15) / 16);

    // 1) p = z @ W^T + b  (WMMA f32)
    gemm_wmma_f32<<<gGemm, 32, 0, stream>>>(z, W, b, p, B, C, D);
    // 2) batch entropy + argmax
    ent_argmax_kernel<<<B, 256, 0, stream>>>(p, E_all, y_all, C, N);
    // 3) support classes + entropies
    support_argmax_kernel<<<(N + 7) / 8, 256, 0, stream>>>(labels, ent, y_all, E_all, N, C);
    // 4) histogram -> offsets
    hipMemsetAsync(counts, 0, sizeof(int) * (size_t)C, stream);
    count_kernel<<<(Ntot + 255) / 256, 256, 0, stream>>>(y_all, counts, Ntot);
    scan_kernel<<<1, 1, 0, stream>>>(counts, offs, C);
    // 5) deterministic class-ordered compaction
    compact_kernel<<<C, 32, 0, stream>>>(y_all, offs, idxArr, Ntot);
    // 6) per-class top-filter_K by lowest entropy (stable)
    select_kernel<<<C, 256, 0, stream>>>(offs, counts, idxArr, E_all, fk, keepM);
    // 7) prototype weights, column-normalized, as Wn[C,D]
    proto_kernel<<<C, 256, 0, stream>>>(offs, counts, idxArr, keepM, supports, z, Wn, N, D);
    // 8) out = z @ Wn^T  (WMMA f32)
    gemm_wmma_f32<<<gGemm, 32, 0, stream>>>(z, Wn, nullptr, (float*)d_out, B, C, D);
}